// CausalDiscovery_54494545051777
// MI455X (gfx1250) — compile-verified
//
#include <hip/hip_runtime.h>

typedef float v2f __attribute__((ext_vector_type(2)));
typedef float v8f __attribute__((ext_vector_type(8)));

#define DD 256
#define BB 32
#define HH 64

// ---------------------------------------------------------------------------
// Part A: adj[b,i,j] = sigmoid( sum_h silu(z_i*W1[0,h]+z_j*W1[1,h]+b1[h])*W2[h] + b2 ) * (i!=j)
// One block per (b,i) row; thread j in [0,256). Weights cached in LDS.
// ---------------------------------------------------------------------------
__global__ __launch_bounds__(256) void adj_kernel(
    const float* __restrict__ z, const float* __restrict__ W1,
    const float* __restrict__ b1, const float* __restrict__ W2,
    const float* __restrict__ b2, float* __restrict__ out)
{
    __shared__ float s0[HH], s1[HH], sb1[HH], sW2[HH];
    int t = threadIdx.x;
    if (t < HH) {
        s0[t]  = W1[t];        // W1[0,h]
        s1[t]  = W1[HH + t];   // W1[1,h]
        sb1[t] = b1[t];
        sW2[t] = W2[t];
    }
    __syncthreads();

    int b = blockIdx.x >> 8;     // 0..31
    int i = blockIdx.x & 255;    // 0..255
    int j = t;                   // 0..255

    float zi = z[b * DD + i];
    float zj = z[b * DD + j];

    float acc = 0.0f;
#pragma unroll 8
    for (int h = 0; h < HH; ++h) {
        float arg = fmaf(zi, s0[h], fmaf(zj, s1[h], sb1[h]));
        float sg  = 1.0f / (1.0f + __expf(-arg));     // sigmoid(arg)
        acc = fmaf(arg * sg, sW2[h], acc);            // silu(arg)*W2[h]
    }
    float logit = acc + b2[0];
    float val = (i == j) ? 0.0f : (1.0f / (1.0f + __expf(-logit)));
    out[(size_t)blockIdx.x * DD + j] = val;
}

// ---------------------------------------------------------------------------
// Part B prep: A0 = (sigmoid(ew) * offdiag)^2 / 2^8 ;  P = I
// ---------------------------------------------------------------------------
__global__ __launch_bounds__(256) void prep_kernel(
    const float* __restrict__ ew, float* __restrict__ A0, float* __restrict__ P)
{
    int idx = blockIdx.x * 256 + threadIdx.x;   // 0..65535
    int i = idx >> 8, j = idx & 255;
    float w = 1.0f / (1.0f + __expf(-ew[idx]));
    float m = (i == j) ? 0.0f : w * w;
    A0[idx] = m * (1.0f / 256.0f);              // scale by 2^-8
    P[idx]  = (i == j) ? 1.0f : 0.0f;
}

// ---------------------------------------------------------------------------
// 256x256x256 FP32 GEMM via V_WMMA_F32_16X16X4_F32.
// C = alpha * (A @ B) + addIdentity * I
// One wave per 16x16 C tile; 8 waves/block, 32 blocks -> 256 tiles.
// Fragment layouts per CDNA5 ISA 7.12.2:
//   A 16x4 : a[v] = A[m=tm+l16][k0 + v + 2*half]
//   B 4x16 : b[v] = B[k0 + v + 2*half][n=tn+l16]
//   C 16x16: acc[r] = C[tm + r + 8*half][tn + l16]
// ---------------------------------------------------------------------------
__global__ __launch_bounds__(256) void gemm_wmma(
    const float* __restrict__ A, const float* __restrict__ Bm,
    float* __restrict__ C, float alpha, int addIdentity)
{
    int wave = threadIdx.x >> 5;          // 0..7
    int lane = threadIdx.x & 31;
    int l16  = lane & 15;
    int half = lane >> 4;

    int tileId = blockIdx.x * 8 + wave;   // 0..255
    int tm = (tileId >> 4) * 16;
    int tn = (tileId & 15) * 16;

    const float* arow = A + (size_t)(tm + l16) * DD;

    v8f acc = {};
#pragma unroll 4
    for (int k0 = 0; k0 < DD; k0 += 4) {
        int kk = k0 + 2 * half;
        v2f a, b;
        a.x = arow[kk];
        a.y = arow[kk + 1];
        b.x = Bm[(size_t)kk * DD + tn + l16];
        b.y = Bm[(size_t)(kk + 1) * DD + tn + l16];
        acc = __builtin_amdgcn_wmma_f32_16x16x4_f32(
            /*neg_a=*/false, a, /*neg_b=*/false, b,
            /*c_mod=*/(short)0, acc, /*reuse_a=*/false, /*reuse_b=*/false);
    }

#pragma unroll
    for (int r = 0; r < 8; ++r) {
        int row = tm + r + 8 * half;
        int col = tn + l16;
        float v = acc[r] * alpha;
        if (addIdentity && row == col) v += 1.0f;
        C[(size_t)row * DD + col] = v;
    }
}

// ---------------------------------------------------------------------------
// trace(E) - 256 -> out[0]
// ---------------------------------------------------------------------------
__global__ __launch_bounds__(256) void trace_kernel(
    const float* __restrict__ E, float* __restrict__ out)
{
    __shared__ float red[256];
    int t = threadIdx.x;
    red[t] = E[(size_t)t * DD + t];
    __syncthreads();
    for (int s = 128; s > 0; s >>= 1) {
        if (t < s) red[t] += red[t + s];
        __syncthreads();
    }
    if (t == 0) out[0] = red[0] - (float)DD;
}

// ---------------------------------------------------------------------------
extern "C" void kernel_launch(void* const* d_in, const int* in_sizes, int n_in,
                              void* d_out, int out_size, void* d_ws, size_t ws_size,
                              hipStream_t stream)
{
    (void)in_sizes; (void)n_in; (void)out_size; (void)ws_size;
    const float* z  = (const float*)d_in[0];
    const float* ew = (const float*)d_in[1];
    const float* W1 = (const float*)d_in[2];
    const float* b1 = (const float*)d_in[3];
    const float* W2 = (const float*)d_in[4];
    const float* b2 = (const float*)d_in[5];
    float* out = (float*)d_out;

    // Output 1: adj (32*256*256 floats at out[0..])
    adj_kernel<<<BB * DD, 256, 0, stream>>>(z, W1, b1, W2, b2, out);

    // Output 2: h = trace(expm(M)) - 256, scaling-and-squaring in d_ws
    float* ws = (float*)d_ws;
    float* A0 = ws;                 // 256KB
    float* P  = ws + DD * DD;       // 256KB
    float* T  = ws + 2 * DD * DD;   // 256KB

    prep_kernel<<<(DD * DD) / 256, 256, 0, stream>>>(ew, A0, P);

    // Horner: P <- I + (A0/k) @ P, k = 12..1   => P ~= exp(A0)
    for (int k = 12; k >= 1; --k) {
        gemm_wmma<<<32, 256, 0, stream>>>(A0, P, T, 1.0f / (float)k, 1);
        float* tmp = P; P = T; T = tmp;
    }
    // Squaring: 8 times => exp(A0 * 2^8) = exp(M)
    for (int s = 0; s < 8; ++s) {
        gemm_wmma<<<32, 256, 0, stream>>>(P, P, T, 1.0f, 0);
        float* tmp = P; P = T; T = tmp;
    }

    trace_kernel<<<1, 256, 0, stream>>>(P, out + (size_t)BB * DD * DD);
}